// ModelNew_25056839205169
// MI455X (gfx1250) — compile-verified
//
#include <hip/hip_runtime.h>
#include <cstdint>

// Problem constants (match reference setup_inputs()):
#define N_  8
#define C_  64
#define D_  16
#define HW_ 16384           // H*W = 128*128
#define PXB 128             // pixels (and threads) per block
#define DEPTH 5             // ring-buffer slots (>= AHEAD + 2)
#define AHEAD 3             // slabs kept in flight
#define SCALING 2.0f

// ---- CDNA5 async-counter wait -------------------------------------------
#if __has_builtin(__builtin_amdgcn_s_wait_asynccnt)
#define WAIT_ASYNC(n) __builtin_amdgcn_s_wait_asynccnt(n)
#else
#define WAIT_ASYNC(n) asm volatile("s_wait_asynccnt %0" :: "n"(n) : "memory")
#endif

// Issue one channel-slab (16 x 128 f32) into LDS via async global->LDS DMA.
// Wave w loads depths 4w..4w+3; each lane moves 16B (4 px), one b128 per depth
// => 4 async ops per wave per slab, each covering a full 512B contiguous row.
__device__ __forceinline__ void issue_slab(const float* __restrict__ xb,
                                           float (&slab)[DEPTH][D_][PXB],
                                           int buf, int wave, int lane) {
#pragma unroll
  for (int j = 0; j < 4; ++j) {
    const int d = wave * 4 + j;
    const float* gp = xb + (unsigned)d * (unsigned)HW_ + (unsigned)(lane * 4);
    // Low 32 bits of a generic pointer to __shared__ are the wave-relative
    // LDS byte offset (ISA 10.2 aperture mapping) -> exactly what VDST wants.
    unsigned ldsa = (unsigned)(uintptr_t)(&slab[buf][d][lane * 4]);
    asm volatile("global_load_async_to_lds_b128 %0, %1, off"
                 :: "v"(ldsa), "v"(gp)
                 : "memory");
  }
}

__global__ __launch_bounds__(PXB) void
meanD_softmaxC_tanh_kernel(const float* __restrict__ x,
                           const float* __restrict__ bias,
                           float* __restrict__ out) {
  __shared__ float slab[DEPTH][D_][PXB];  // 5-deep ring of channel slabs
  __shared__ float logits[C_ * PXB];      // per-pixel channel logits

  const int tid  = threadIdx.x;
  const int wave = tid >> 5;
  const int lane = tid & 31;

  const unsigned p0  = blockIdx.x * (unsigned)PXB; // first pixel of block
  const unsigned n   = p0 >> 14;                   // / (H*W)
  const unsigned hw0 = p0 & (HW_ - 1u);
  const unsigned nc  = n * (unsigned)C_;

  // Prologue: put AHEAD slabs in flight (channels 0..AHEAD-1).
#pragma unroll
  for (int c = 0; c < AHEAD; ++c) {
    issue_slab(x + ((size_t)(nc + c) * D_ * HW_ + hw0), slab, c, wave, lane);
  }

  int cur = 0;      // ring slot holding slab c
  int nxt = AHEAD;  // ring slot for slab c+AHEAD
  for (int c = 0; c < C_; ++c) {
    // Keep AHEAD slabs in flight; ring depth AHEAD+2 means the slot we issue
    // into was consumed at iteration c-2, already ordered by the barrier of
    // iteration c-1 -> only ONE barrier per iteration is needed.
    if (c + AHEAD < C_) {
      issue_slab(x + ((size_t)(nc + c + AHEAD) * D_ * HW_ + hw0),
                 slab, nxt, wave, lane);
      WAIT_ASYNC(4 * AHEAD);            // oldest (slab c) has landed
    } else if (c == C_ - 3) { WAIT_ASYNC(8); }
    else if (c == C_ - 2)   { WAIT_ASYNC(4); }
    else                    { WAIT_ASYNC(0); }
    __syncthreads();                    // slab c visible from all waves

    float s = 0.f;
#pragma unroll
    for (int d = 0; d < D_; ++d) s += slab[cur][d][tid];
    logits[c * PXB + tid] = fmaf(s, 1.0f / (float)D_, bias[c]);

    cur = (cur == DEPTH - 1) ? 0 : cur + 1;
    nxt = (nxt == DEPTH - 1) ? 0 : nxt + 1;
  }

  // ---- per-pixel softmax over C + tanh; thread owns its column, no syncs ----
  float m = logits[tid];
#pragma unroll 8
  for (int c = 1; c < C_; ++c) m = fmaxf(m, logits[c * PXB + tid]);

  float ssum = 0.f;
#pragma unroll 8
  for (int c = 0; c < C_; ++c) {
    const float e = __expf(logits[c * PXB + tid] - m);
    logits[c * PXB + tid] = e;
    ssum += e;
  }
  const float inv = 1.0f / ssum;

  float* op = out + (size_t)nc * HW_ + hw0 + tid;
#pragma unroll 4
  for (int c = 0; c < C_; ++c) {
    const float p = logits[c * PXB + tid] * inv;
#if __has_builtin(__builtin_amdgcn_tanhf)
    const float t = __builtin_amdgcn_tanhf(p);
#else
    const float e2 = __expf(2.0f * p);
    const float t  = (e2 - 1.0f) / (e2 + 1.0f);
#endif
    __builtin_nontemporal_store(SCALING * t, op + (size_t)c * HW_);
  }
}

extern "C" void kernel_launch(void* const* d_in, const int* in_sizes, int n_in,
                              void* d_out, int out_size, void* d_ws, size_t ws_size,
                              hipStream_t stream) {
  (void)in_sizes; (void)n_in; (void)out_size; (void)d_ws; (void)ws_size;
  const float* x    = (const float*)d_in[0];
  const float* bias = (const float*)d_in[1];
  float*       out  = (float*)d_out;

  dim3 grid((N_ * HW_) / PXB);  // 1024 blocks
  dim3 block(PXB);              // 128 threads = 4 waves (wave32)
  hipLaunchKernelGGL(meanD_softmaxC_tanh_kernel, grid, block, 0, stream,
                     x, bias, out);
}